// RBFN_74586402063116
// MI455X (gfx1250) — compile-verified
//
#include <hip/hip_runtime.h>

// RBFN fused kernel for MI455X (gfx1250, wave32).
// Phase 1: L1 distances + exp (VALU-bound, dominant cost) staged through LDS.
// Phase 2: radial @ W^T + b via V_WMMA_F32_16X16X4_F32 (f32-native matrix op),
//          branchless B-padding (mask-multiply, no exec divergence in loop).

#define N_INPUT     4096
#define NUM_CENTERS 512
#define DIM         128
#define N_OUT       10

#define TILE_ROWS   16
#define BLOCK       256
// Padding: natural strides 128 and 512 are multiples of 64 banks -> full
// bank conflicts. 132 (=4 mod 64, float4-aligned) and 513 (=1 mod 64) fix it.
#define BATCH_STRIDE 132
#define RAD_STRIDE   513

typedef __attribute__((ext_vector_type(2))) float v2f;
typedef __attribute__((ext_vector_type(8))) float v8f;

__global__ __launch_bounds__(BLOCK)
void rbfn_fused_kernel(const float* __restrict__ batches,
                       const float* __restrict__ centers,
                       const float* __restrict__ sigma_p,
                       const float* __restrict__ W,
                       const float* __restrict__ bias,
                       float* __restrict__ out)
{
    __shared__ float batch_lds[TILE_ROWS * BATCH_STRIDE];   // 8448 B
    __shared__ float radial_lds[TILE_ROWS * RAD_STRIDE];    // 32832 B

    const int tid  = threadIdx.x;
    const int row0 = blockIdx.x * TILE_ROWS;
    const float sigma = sigma_p[0];   // uniform -> s_load

    // ---- Phase 0: stage 16x128 batch tile into LDS (coalesced float4) ----
    {
        const float4* gsrc = (const float4*)(batches + (size_t)row0 * DIM);
        for (int i = tid; i < TILE_ROWS * (DIM / 4); i += BLOCK) {
            const int r  = i >> 5;      // / (DIM/4)
            const int c4 = i & 31;
            float4 v = gsrc[r * (DIM / 4) + c4];
            float* dst = &batch_lds[r * BATCH_STRIDE + c4 * 4];
            dst[0] = v.x; dst[1] = v.y; dst[2] = v.z; dst[3] = v.w;
        }
    }
    __syncthreads();

    // ---- Phase 1: L1 distance + exp(-sigma*dist) -> radial_lds ----
    // Thread (r = tid&15, cgrp = tid>>4) handles centers c = 16*i + cgrp.
    // Center rows are shared by 16 lanes -> broadcast-friendly vmem loads.
    {
        const int r    = tid & 15;
        const int cgrp = tid >> 4;
        const float* brow = &batch_lds[r * BATCH_STRIDE];
        for (int i = 0; i < NUM_CENTERS / 16; ++i) {
            const int c = i * 16 + cgrp;
            const float4* crow = (const float4*)(centers + (size_t)c * DIM);
            float acc = 0.0f;
#pragma unroll 8
            for (int d4 = 0; d4 < DIM / 4; ++d4) {
                float4 cv = crow[d4];
                acc += __builtin_fabsf(brow[d4 * 4 + 0] - cv.x);
                acc += __builtin_fabsf(brow[d4 * 4 + 1] - cv.y);
                acc += __builtin_fabsf(brow[d4 * 4 + 2] - cv.z);
                acc += __builtin_fabsf(brow[d4 * 4 + 3] - cv.w);
            }
            radial_lds[r * RAD_STRIDE + c] = __expf(-sigma * acc);
        }
    }
    __syncthreads();

    // ---- Phase 2: wave 0 computes the 16x16 (N_OUT=10 padded) GEMM tile
    //      over K=512 with chained V_WMMA_F32_16X16X4_F32 ops. ----
    if (tid < 32) {
        const int lane = tid;
        const int half = lane >> 4;   // 0: K = k..k+1, 1: K = k+2..k+3
        const int m    = lane & 15;   // A-row (lanes) / B-col (lanes)

        // Branchless zero-padding for B columns >= N_OUT:
        //  - safe pointer (lanes m>=10 alias row 0, value discarded by mask)
        //  - loop-invariant multiplicative mask -> v_mul, no exec toggling
        const bool  valid = (m < N_OUT);
        const float mask  = valid ? 1.0f : 0.0f;
        const float* wrow = W + (size_t)(valid ? m : 0) * NUM_CENTERS;
        const float* arow = radial_lds + m * RAD_STRIDE;

        v8f acc0 = {};
        v8f acc1 = {};
        for (int k = 0; k < NUM_CENTERS; k += 8) {
            const int kk0 = k + 2 * half;          // even -> 8B aligned
            const int kk1 = k + 4 + 2 * half;

            v2f a0, a1, b0, b1;
            // A fragments: 16x4 f32 layout (ISA 7.12.2), from LDS
            a0.x = arow[kk0];
            a0.y = arow[kk0 + 1];
            a1.x = arow[kk1];
            a1.y = arow[kk1 + 1];
            // B fragments: Wt[k][n] = W[n][k], one global_load_b64 each
            const float2 w0 = *(const float2*)(wrow + kk0);
            const float2 w1 = *(const float2*)(wrow + kk1);
            b0.x = w0.x * mask;  b0.y = w0.y * mask;
            b1.x = w1.x * mask;  b1.y = w1.y * mask;

            acc0 = __builtin_amdgcn_wmma_f32_16x16x4_f32(
                false, a0, false, b0, (short)0, acc0, false, false);
            acc1 = __builtin_amdgcn_wmma_f32_16x16x4_f32(
                false, a1, false, b1, (short)0, acc1, false, false);
        }

        // D layout: VGPR v -> M = v + 8*half (lanes split), N = m.
        if (valid) {
            const float bv = bias[m];
#pragma unroll
            for (int v = 0; v < 8; ++v) {
                const int row = row0 + v + 8 * half;
                out[(size_t)row * N_OUT + m] = acc0[v] + acc1[v] + bv;
            }
        }
    }
}

extern "C" void kernel_launch(void* const* d_in, const int* in_sizes, int n_in,
                              void* d_out, int out_size, void* d_ws, size_t ws_size,
                              hipStream_t stream) {
    const float* batches = (const float*)d_in[0];   // [4096,128]
    const float* centers = (const float*)d_in[1];   // [512,128]
    const float* sigma   = (const float*)d_in[2];   // scalar
    const float* W       = (const float*)d_in[3];   // [10,512]
    const float* bias    = (const float*)d_in[4];   // [10]
    float* out           = (float*)d_out;           // [4096,10]

    (void)in_sizes; (void)n_in; (void)out_size; (void)d_ws; (void)ws_size;

    rbfn_fused_kernel<<<N_INPUT / TILE_ROWS, BLOCK, 0, stream>>>(
        batches, centers, sigma, W, bias, out);
}